// MetaNet2_16355235463653
// MI455X (gfx1250) — compile-verified
//
#include <hip/hip_runtime.h>
#include <cmath>

typedef __attribute__((ext_vector_type(2))) float v2f;
typedef __attribute__((ext_vector_type(8))) float v8f;

#define NBATCH   64
#define W_TOTAL_ 2234368
#define B_TOTAL_ 3526

// One wave computes a 16-wide output tile of one sample's layer:
//   out[o0..o0+15] = sum_k h[k] * W[k, o0..o0+15]  (+bias, +activation)
// via V_WMMA_F32_16X16X4_F32 with the A matrix's 16 rows all holding h
// (every D row is the same answer; only the W loads touch HBM, each weight
// is read exactly once -> bandwidth bound at 23.3 TB/s).
__global__ __launch_bounds__(256) void metanet_layer(
    const float* __restrict__ Wall,   // (64, W_TOTAL) fp32
    const float* __restrict__ Ball,   // (64, B_TOTAL) fp32
    const float* __restrict__ hin,    // (64, din)
    float*       __restrict__ hout,   // (64, dout)
    int din, int dout, int w_off, int b_off, int act)
{
    __shared__ float hs[1872];        // max din_pad16 = 1872 (for din = 1862)

    const int b    = blockIdx.y;
    const int tid  = threadIdx.x;
    const int lane = tid & 31;
    const int wave = tid >> 5;

    // Stage h into LDS, zero-padded to a multiple of 16 so the K loop is
    // unmasked and can run 4 WMMAs per trip with no tail.
    const int din_pad = (din + 15) & ~15;
    const float* hptr = hin + (size_t)b * din;
    for (int i = tid; i < din_pad; i += 256)
        hs[i] = (i < din) ? hptr[i] : 0.0f;
    __syncthreads();

    const int o0 = (blockIdx.x * 8 + wave) * 16;
    if (o0 >= dout) return;           // wave-uniform: EXEC stays all-ones for WMMA

    const int oc    = o0 + (lane & 15);
    const int ocl   = (oc < dout) ? oc : (dout - 1);   // clamp tail cols (last layer)
    const int khalf = (lane >> 4) << 1;                // lanes 0-15 -> K+0/1, 16-31 -> K+2/3

    const float* Wb = Wall + (size_t)b * W_TOTAL_ + (size_t)w_off;

    // Loop-carried row pointer: rows (k+khalf, k+khalf+1) live at r, r+dout.
    // Advance by 4 rows per WMMA -- pure pointer adds, no per-iter 64-bit mul.
    const float* r = Wb + (size_t)khalf * dout + ocl;
    const size_t step4 = (size_t)4 * dout;

    v8f c0 = {};
    v8f c1 = {};

    for (int k = 0; k < din_pad; k += 16) {
        const float* ra = r;
        const float* rb = ra + step4;
        const float* rc = rb + step4;
        const float* rd = rc + step4;
        r = rd + step4;

        v2f a0, a1, a2, a3, bm0, bm1, bm2, bm3;
        a0.x = hs[k +      khalf];  a0.y = hs[k +      khalf + 1];
        a1.x = hs[k +  4 + khalf];  a1.y = hs[k +  4 + khalf + 1];
        a2.x = hs[k +  8 + khalf];  a2.y = hs[k +  8 + khalf + 1];
        a3.x = hs[k + 12 + khalf];  a3.y = hs[k + 12 + khalf + 1];

        bm0.x = ra[0];  bm0.y = ra[dout];
        bm1.x = rb[0];  bm1.y = rb[dout];
        bm2.x = rc[0];  bm2.y = rc[dout];
        bm3.x = rd[0];  bm3.y = rd[dout];

        // 8 args: (neg_a, A, neg_b, B, c_mod, C, reuse_a, reuse_b)
        c0 = __builtin_amdgcn_wmma_f32_16x16x4_f32(false, a0, false, bm0,
                                                   (short)0, c0, false, false);
        c1 = __builtin_amdgcn_wmma_f32_16x16x4_f32(false, a1, false, bm1,
                                                   (short)0, c1, false, false);
        c0 = __builtin_amdgcn_wmma_f32_16x16x4_f32(false, a2, false, bm2,
                                                   (short)0, c0, false, false);
        c1 = __builtin_amdgcn_wmma_f32_16x16x4_f32(false, a3, false, bm3,
                                                   (short)0, c1, false, false);
    }

    // Every D row is identical; lanes 0-15 of VGPR0 hold N = 0..15.
    float v = c0[0] + c1[0];

    if ((lane < 16) && (oc < dout)) {
        v += Ball[(size_t)b * B_TOTAL_ + b_off + oc];
        float res;
        if (act == 0) res = v / (1.0f + __expf(-v));   // SiLU
        else          res = tanhf(v);                  // final tanh
        hout[(size_t)b * dout + oc] = res;
    }
}

extern "C" void kernel_launch(void* const* d_in, const int* in_sizes, int n_in,
                              void* d_out, int out_size, void* d_ws, size_t ws_size,
                              hipStream_t stream) {
    const float* x = (const float*)d_in[0];
    const float* W = (const float*)d_in[1];
    const float* B = (const float*)d_in[2];
    float* out = (float*)d_out;

    // Ping-pong activation buffers in scratch (max h size = 64*512 floats).
    float* hA = (float*)d_ws;
    float* hB = hA + NBATCH * 512;

    static const int dims[7] = {1862, 512, 256, 128, 256, 512, 1862};

    int w_off = 0, b_off = 0;
    const float* cur_in = x;
    for (int i = 0; i < 6; ++i) {
        const int din  = dims[i];
        const int dout = dims[i + 1];
        float* cur_out = (i == 5) ? out : ((i % 2 == 0) ? hA : hB);

        dim3 grid((dout + 127) / 128, NBATCH);
        metanet_layer<<<grid, 256, 0, stream>>>(W, B, cur_in, cur_out,
                                                din, dout, w_off, b_off,
                                                (i < 5) ? 0 : 1);
        cur_in = cur_out;
        w_off += din * dout;
        b_off += dout;
    }
}